// GCN_LSTM_Probabilistic_11510512353640
// MI455X (gfx1250) — compile-verified
//
#include <hip/hip_runtime.h>
#include <hip/hip_bf16.h>
#include <math.h>

typedef __attribute__((ext_vector_type(2))) float v2f;
typedef __attribute__((ext_vector_type(8))) float v8f;

#define NUM_GRAPHS 64
#define FDIM 128
#define HDIM 256
#define NGATES 1024

// GEMM tiling
#define TM 64            // rows per block
#define KC 32            // K-chunk staged in LDS
#define SXLD 132         // padded LDS row stride (floats): 132 % 64 == 4 -> conflict-free A reads

static __device__ __forceinline__ float sigmoidf_(float x) {
  return 1.0f / (1.0f + expf(-x));
}

// ---------------- utility fills ----------------
__global__ void fill_kernel(float* __restrict__ p, float v, int n) {
  int i = blockIdx.x * blockDim.x + threadIdx.x;
  if (i < n) p[i] = v;
}

// ---------------- degree / dinv ----------------
__global__ void deg_accum_kernel(const int* __restrict__ dst, float* __restrict__ deg, int E) {
  int i = blockIdx.x * blockDim.x + threadIdx.x;
  if (i < E) atomicAdd(&deg[dst[i]], 1.0f);
}

__global__ void rsqrt_kernel(float* __restrict__ deg, int n) {
  int i = blockIdx.x * blockDim.x + threadIdx.x;
  if (i < n) deg[i] = rsqrtf(deg[i]);   // deg >= 1 (self-loop), no zero case
}

// ---------------- fp32 WMMA GEMM:  Y[M x 128] = X[M x 128] * W[128 x 128] ----------------
// Block = 256 threads = 8 waves. Each block computes a TM x 128 output tile:
// wave w owns column tile w (16 cols) and iterates 4 row tiles (TM/16).
// X tile staged in LDS (zero-padded tail -> no divergence in the WMMA loop),
// W staged in KC-wide K-chunks. All WMMA operands come from LDS.
__global__ void gemm128_wmma_f32(const float* __restrict__ X,
                                 const float* __restrict__ W,
                                 float* __restrict__ Y, int M) {
  __shared__ float sX[TM * SXLD];      // 64 rows, padded stride   (~33.8 KB)
  __shared__ float sW[KC * FDIM];      // K-chunk of W             (16 KB)

  const int tid  = threadIdx.x;
  const int row0 = blockIdx.x * TM;

  // ---- cooperative load of X tile (float4, zero-pad rows >= M) ----
  {
    const float4* X4 = (const float4*)X;
    float4* sX4 = (float4*)sX;                  // row r starts at float4 index r*(SXLD/4)
    for (int i = tid; i < TM * (FDIM / 4); i += 256) {
      int r  = i >> 5;                          // / (FDIM/4)
      int c4 = i & 31;
      int gr = row0 + r;
      float4 v = make_float4(0.f, 0.f, 0.f, 0.f);
      if (gr < M) v = X4[(size_t)gr * (FDIM / 4) + c4];
      sX4[r * (SXLD / 4) + c4] = v;
    }
  }

  const int lane    = tid & 31;
  const int wave    = tid >> 5;                 // 0..7 -> column tile
  const int halfsel = lane >> 4;                // 0: lanes 0-15, 1: lanes 16-31
  const int l       = lane & 15;
  const int colB    = wave * 16 + l;            // B/D column held by this lane

  v8f acc[TM / 16];
#pragma unroll
  for (int rt = 0; rt < TM / 16; ++rt)
    acc[rt] = v8f{0.f, 0.f, 0.f, 0.f, 0.f, 0.f, 0.f, 0.f};

  const float4* W4 = (const float4*)W;
  for (int kc = 0; kc < FDIM; kc += KC) {
    // ---- stage W[kc .. kc+KC) x 128 ----
    __syncthreads();
    {
      float4* sW4 = (float4*)sW;
      for (int i = tid; i < KC * (FDIM / 4); i += 256)
        sW4[i] = W4[(size_t)kc * (FDIM / 4) + i];
    }
    __syncthreads();

#pragma unroll
    for (int k = 0; k < KC; k += 4) {
      const int ka = k + (halfsel << 1);        // VGPR0 holds K=k / K=k+2 halves
      v2f b;
      b.x = sW[ka * FDIM + colB];
      b.y = sW[(ka + 1) * FDIM + colB];
#pragma unroll
      for (int rt = 0; rt < TM / 16; ++rt) {
        const float* xr = sX + (rt * 16 + l) * SXLD + kc + ka;
        v2f a;
        a.x = xr[0];
        a.y = xr[1];
        acc[rt] = __builtin_amdgcn_wmma_f32_16x16x4_f32(false, a, false, b,
                                                        (short)0, acc[rt],
                                                        false, false);
      }
    }
  }

  // ---- store: D layout: VGPR v -> M = v (lanes 0-15) or v+8 (lanes 16-31) ----
#pragma unroll
  for (int rt = 0; rt < TM / 16; ++rt) {
    int rbase = row0 + rt * 16 + (halfsel << 3);
#pragma unroll
    for (int v = 0; v < 8; ++v) {
      int row = rbase + v;
      if (row < M) Y[(size_t)row * FDIM + colB] = acc[rt][v];
    }
  }
}

// ---------------- GCN aggregation ----------------
// out[v] = H[v] * dinv[v]^2   (self-loop term)
__global__ void agg_init_kernel(const float* __restrict__ H, const float* __restrict__ dinv,
                                float* __restrict__ out, int total) {
  int i = blockIdx.x * blockDim.x + threadIdx.x;
  if (i >= total) return;
  float dv = dinv[i >> 7];              // / FDIM
  out[i] = H[i] * dv * dv;
}

// out[dst] += H[src] * dinv[src]*dinv[dst]; one edge per 32 lanes, float4 per lane
__global__ void agg_edges_kernel(const float* __restrict__ H, const int* __restrict__ src,
                                 const int* __restrict__ dst, const float* __restrict__ dinv,
                                 float* __restrict__ out, int E) {
  int gid  = blockIdx.x * blockDim.x + threadIdx.x;
  int e    = gid >> 5;
  int lane = gid & 31;
  if (e >= E) return;
  int s = src[e], d = dst[e];
  float norm = dinv[s] * dinv[d];
  const float4* hs = (const float4*)(H + (size_t)s * FDIM);
  float4 hv = hs[lane];
  float* po = out + (size_t)d * FDIM + lane * 4;
  atomicAdd(po + 0, hv.x * norm);
  atomicAdd(po + 1, hv.y * norm);
  atomicAdd(po + 2, hv.z * norm);
  atomicAdd(po + 3, hv.w * norm);
}

__global__ void bias_relu_kernel(float* __restrict__ out, const float* __restrict__ b, int total) {
  int i = blockIdx.x * blockDim.x + threadIdx.x;
  if (i >= total) return;
  float v = out[i] + b[i & (FDIM - 1)];
  out[i] = v > 0.f ? v : 0.f;
}

// ---------------- global mean pool ----------------
__global__ void pool_accum_kernel(const float* __restrict__ H, const int* __restrict__ batch,
                                  float* __restrict__ sums, float* __restrict__ cnts, int N) {
  int gid  = blockIdx.x * blockDim.x + threadIdx.x;
  int v    = gid >> 5;
  int lane = gid & 31;
  if (v >= N) return;
  int g = batch[v];
  if (lane == 0) atomicAdd(&cnts[g], 1.0f);
  const float4* hv = (const float4*)(H + (size_t)v * FDIM);
  float4 h = hv[lane];
  float* ps = sums + (size_t)g * FDIM + lane * 4;
  atomicAdd(ps + 0, h.x);
  atomicAdd(ps + 1, h.y);
  atomicAdd(ps + 2, h.z);
  atomicAdd(ps + 3, h.w);
}

__global__ void pool_div_kernel(float* __restrict__ sums, const float* __restrict__ cnts) {
  int i = blockIdx.x * blockDim.x + threadIdx.x;
  if (i >= NUM_GRAPHS * FDIM) return;
  sums[i] /= fmaxf(cnts[i >> 7], 1.0f);
}

// ---------------- LSTM step (h0 = c0 = 0) ----------------
__global__ void lstm_gates_kernel(const float* __restrict__ pooled,
                                  const float* __restrict__ W_ih,
                                  const float* __restrict__ b_ih,
                                  const float* __restrict__ b_hh,
                                  float* __restrict__ gates) {
  int idx = blockIdx.x * blockDim.x + threadIdx.x;
  if (idx >= NUM_GRAPHS * NGATES) return;
  int g = idx >> 10, j = idx & (NGATES - 1);
  float acc = b_ih[j] + b_hh[j];
  const float* xr = pooled + (size_t)g * FDIM;
  const float* wr = W_ih + (size_t)j * FDIM;
#pragma unroll 4
  for (int k = 0; k < FDIM; ++k) acc += xr[k] * wr[k];
  gates[idx] = acc;
}

__global__ void lstm_act_kernel(const float* __restrict__ gates, float* __restrict__ hT) {
  int idx = blockIdx.x * blockDim.x + threadIdx.x;
  if (idx >= NUM_GRAPHS * HDIM) return;
  int g = idx >> 8, h = idx & (HDIM - 1);
  const float* gr = gates + (size_t)g * NGATES;
  float ig = sigmoidf_(gr[h]);              // i gate
  float gg = tanhf(gr[2 * HDIM + h]);       // g gate  (f*c0 = 0)
  float og = sigmoidf_(gr[3 * HDIM + h]);   // o gate
  float cc = ig * gg;
  hT[idx] = og * tanhf(cc);
}

// ---------------- heads: mean, softplus(log_var) ----------------
__global__ void head_kernel(const float* __restrict__ hT, const float* __restrict__ Wm,
                            const float* __restrict__ bm, const float* __restrict__ Wv,
                            const float* __restrict__ bv, float* __restrict__ out) {
  int g = threadIdx.x;
  if (g >= NUM_GRAPHS) return;
  const float* hr = hT + (size_t)g * HDIM;
  float m = 0.f, vv = 0.f;
  for (int k = 0; k < HDIM; ++k) {
    m  += hr[k] * Wm[k];
    vv += hr[k] * Wv[k];
  }
  m  += bm[0];
  vv += bv[0];
  out[g] = m;
  out[NUM_GRAPHS + g] = fmaxf(vv, 0.f) + log1pf(expf(-fabsf(vv)));  // stable softplus
}

// ---------------- launch ----------------
extern "C" void kernel_launch(void* const* d_in, const int* in_sizes, int n_in,
                              void* d_out, int out_size, void* d_ws, size_t ws_size,
                              hipStream_t stream) {
  const int N = in_sizes[0] / FDIM;
  const int E = in_sizes[1] / 2;
  const int total = N * FDIM;

  const float* x     = (const float*)d_in[0];
  const int*   ei    = (const int*)d_in[1];
  const int*   src   = ei;
  const int*   dst   = ei + E;
  const int*   batch = (const int*)d_in[2];
  const float* W1    = (const float*)d_in[3];
  const float* b1    = (const float*)d_in[4];
  const float* W2    = (const float*)d_in[5];
  const float* b2    = (const float*)d_in[6];
  const float* W_ih  = (const float*)d_in[7];
  // d_in[8] = W_hh: unused, h0 == 0
  const float* b_ih  = (const float*)d_in[9];
  const float* b_hh  = (const float*)d_in[10];
  const float* Wm    = (const float*)d_in[11];
  const float* bm    = (const float*)d_in[12];
  const float* Wv    = (const float*)d_in[13];
  const float* bv    = (const float*)d_in[14];
  float* out = (float*)d_out;

  // workspace carve-up (floats)
  float* ws = (float*)d_ws;
  size_t off = 0;
  float* deg    = ws + off; off += ((size_t)N + 255) & ~(size_t)255;  // becomes dinv in place
  float* bufA   = ws + off; off += (size_t)total;
  float* bufB   = ws + off; off += (size_t)total;
  float* pooled = ws + off; off += (size_t)NUM_GRAPHS * FDIM;
  float* cnts   = ws + off; off += NUM_GRAPHS;                        // contiguous after pooled
  float* gates  = ws + off; off += (size_t)NUM_GRAPHS * NGATES;
  float* hT     = ws + off; off += (size_t)NUM_GRAPHS * HDIM;

  const int B = 256;
  auto cdiv = [](int a, int b) { return (a + b - 1) / b; };
  const int mblocks = cdiv(N, TM);

  // degree / dinv (shared across both GCN layers)
  fill_kernel<<<cdiv(N, B), B, 0, stream>>>(deg, 1.0f, N);            // self-loop
  deg_accum_kernel<<<cdiv(E, B), B, 0, stream>>>(dst, deg, E);
  rsqrt_kernel<<<cdiv(N, B), B, 0, stream>>>(deg, N);

  // layer 1
  gemm128_wmma_f32<<<mblocks, B, 0, stream>>>(x, W1, bufA, N);
  agg_init_kernel<<<cdiv(total, B), B, 0, stream>>>(bufA, deg, bufB, total);
  agg_edges_kernel<<<cdiv(E * 32, B), B, 0, stream>>>(bufA, src, dst, deg, bufB, E);
  bias_relu_kernel<<<cdiv(total, B), B, 0, stream>>>(bufB, b1, total);

  // layer 2
  gemm128_wmma_f32<<<mblocks, B, 0, stream>>>(bufB, W2, bufA, N);
  agg_init_kernel<<<cdiv(total, B), B, 0, stream>>>(bufA, deg, bufB, total);
  agg_edges_kernel<<<cdiv(E * 32, B), B, 0, stream>>>(bufA, src, dst, deg, bufB, E);
  bias_relu_kernel<<<cdiv(total, B), B, 0, stream>>>(bufB, b2, total);

  // global mean pool
  fill_kernel<<<cdiv(NUM_GRAPHS * FDIM + NUM_GRAPHS, B), B, 0, stream>>>(
      pooled, 0.0f, NUM_GRAPHS * FDIM + NUM_GRAPHS);
  pool_accum_kernel<<<cdiv(N * 32, B), B, 0, stream>>>(bufB, batch, pooled, cnts, N);
  pool_div_kernel<<<cdiv(NUM_GRAPHS * FDIM, B), B, 0, stream>>>(pooled, cnts);

  // LSTM step + heads
  lstm_gates_kernel<<<cdiv(NUM_GRAPHS * NGATES, B), B, 0, stream>>>(pooled, W_ih, b_ih, b_hh, gates);
  lstm_act_kernel<<<cdiv(NUM_GRAPHS * HDIM, B), B, 0, stream>>>(gates, hT);
  head_kernel<<<1, NUM_GRAPHS, 0, stream>>>(hT, Wm, bm, Wv, bv, out);
}